// VectorQuantizer_86964497809534
// MI455X (gfx1250) — compile-verified
//
#include <hip/hip_runtime.h>
#include <hip/hip_bf16.h>

typedef __attribute__((ext_vector_type(2))) float v2f;
typedef __attribute__((ext_vector_type(8))) float v8f;

#define N_ROWS 32768
#define E_DIM  256
#define N_E    2048
#define BETA   0.25f

// d_out layout (floats): loss | z_q_st | perplexity | min_encodings | indices
#define OUT_LOSS   0
#define OUT_ZQ     1
#define OUT_PERP   8388609
#define OUT_ONEHOT 8388610
#define OUT_IDX    75497474

// ---------------------------------------------------------------------------
// Kernel 1: per-code squared norms; zero counts + loss accumulator (ws state)
// ---------------------------------------------------------------------------
__global__ void vq_init(const float* __restrict__ emb, float* __restrict__ e2,
                        unsigned int* __restrict__ counts, float* __restrict__ lossAcc) {
    int c = blockIdx.x * blockDim.x + threadIdx.x;
    if (c < N_E) {
        const float* row = emb + (size_t)c * E_DIM;
        float s = 0.f;
#pragma unroll 8
        for (int k = 0; k < E_DIM; ++k) { float v = row[k]; s += v * v; }
        e2[c] = s;
        counts[c] = 0u;
    }
    if (c == 0) *lossAcc = 0.f;
}

// ---------------------------------------------------------------------------
// Kernel 2: WMMA distance GEMM + fused argmin.
// Block = 128 threads (4 waves). Block stages 64 z-rows into 64KB LDS
// (coalesced), each wave hoists its 16x256 A tile into 128 VGPRs
// (v2f areg[64], f32 16x4 A layout) -- invariant across all column tiles.
// Steady-state loop processes TWO 16-code column tiles per iteration with
// two independent accumulators sharing the A registers: doubles the
// independent v_wmma work behind each B-load wait (L2 latency cover),
// tracking argmin of (e^2 - 2 z.e)  (||z||^2 is row-constant -> dropped).
// ---------------------------------------------------------------------------
__global__ __launch_bounds__(128) void vq_argmin_wmma(
    const float* __restrict__ z, const float* __restrict__ emb,
    const float* __restrict__ e2, int* __restrict__ idxOut) {
    __shared__ float ldsZ[64 * E_DIM];  // 64 KB

    const int tid = threadIdx.x;
    const int blockRow = blockIdx.x * 64;

    // Cooperative coalesced stage of 64 rows of z (16384 floats) into LDS
    const float4* zsrc = (const float4*)(z + (size_t)blockRow * E_DIM);
    float4* zdst = (float4*)ldsZ;
#pragma unroll
    for (int i = 0; i < 32; ++i) zdst[i * 128 + tid] = zsrc[i * 128 + tid];
    __syncthreads();

    const int wave = tid >> 5;
    const int lane = tid & 31;
    const int lo = lane & 15;   // A: row M / B,C,D: col N
    const int hi = lane >> 4;   // K-pair select for A/B, +8 row for C/D

    // A-matrix (16x4 f32) lane view: lane holds row M=lo, K = {2*hi, 2*hi+1}.
    const float* aBase = ldsZ + (size_t)(wave * 16 + lo) * E_DIM + 2 * hi;
    v2f areg[64];
#pragma unroll
    for (int kk = 0; kk < 64; ++kk) areg[kk] = *(const v2f*)(aBase + kk * 4);

    float bestD[8];
    int   bestI[8];
#pragma unroll
    for (int r = 0; r < 8; ++r) { bestD[r] = 3.4e38f; bestI[r] = 0; }

    for (int ct = 0; ct < N_E / 32; ++ct) {
        const int colbase = ct * 32;
        // B-matrix (4x16 f32) lane view: lane holds col N=lo (emb row), same K pair
        const float* bBase0 = emb + (size_t)(colbase + lo) * E_DIM + 2 * hi;
        const float* bBase1 = bBase0 + (size_t)16 * E_DIM;

        v8f c0 = {};
        v8f c1 = {};
#pragma unroll
        for (int kk = 0; kk < 64; ++kk) {  // K base = 4*kk
            v2f b0 = *(const v2f*)(bBase0 + kk * 4);
            v2f b1 = *(const v2f*)(bBase1 + kk * 4);
            c0 = __builtin_amdgcn_wmma_f32_16x16x4_f32(
                false, areg[kk], false, b0, (short)0, c0, false, false);
            c1 = __builtin_amdgcn_wmma_f32_16x16x4_f32(
                false, areg[kk], false, b1, (short)0, c1, false, false);
        }

        const int col0 = colbase + lo;
        const int col1 = col0 + 16;
        const float e2c0 = e2[col0];
        const float e2c1 = e2[col1];
        // C/D layout: lane holds (row = r + 8*hi, col = lo) in VGPR r
#pragma unroll
        for (int r = 0; r < 8; ++r) {
            float d0 = e2c0 - 2.0f * c0[r];
            if (d0 < bestD[r]) { bestD[r] = d0; bestI[r] = col0; }  // strict <: first-min
            float d1 = e2c1 - 2.0f * c1[r];
            if (d1 < bestD[r]) { bestD[r] = d1; bestI[r] = col1; }
        }
    }

    // Reduce argmin across the 16 lanes of each half (xor offsets stay in-half)
#pragma unroll
    for (int off = 8; off >= 1; off >>= 1) {
#pragma unroll
        for (int r = 0; r < 8; ++r) {
            float od = __shfl_xor(bestD[r], off, 32);
            int   oi = __shfl_xor(bestI[r], off, 32);
            if (od < bestD[r] || (od == bestD[r] && oi < bestI[r])) {
                bestD[r] = od; bestI[r] = oi;
            }
        }
    }

    if (lo == 0) {
#pragma unroll
        for (int r = 0; r < 8; ++r) {
            int row = blockRow + wave * 16 + r + 8 * hi;
            idxOut[row] = bestI[r];
        }
    }
}

// ---------------------------------------------------------------------------
// Kernel 3: gather z_q, write z_q_st, one-hot scatter, index, counts, loss
// Block = 256 threads = one element per thread per row; 32 rows per block.
// ---------------------------------------------------------------------------
__global__ __launch_bounds__(256) void vq_outputs(
    const float* __restrict__ z, const float* __restrict__ emb,
    const int* __restrict__ idxArr, float* __restrict__ out,
    unsigned int* __restrict__ counts, float* __restrict__ lossAcc) {
    __shared__ float red[256];
    const int t = threadIdx.x;
    const int rowBase = blockIdx.x * 32;

    float acc = 0.f;
    for (int rr = 0; rr < 32; ++rr) {
        int row = rowBase + rr;
        int idx = idxArr[row];
        float zq = emb[(size_t)idx * E_DIM + t];
        float zv = z[(size_t)row * E_DIM + t];
        out[OUT_ZQ + (size_t)row * E_DIM + t] = zq;  // z + sg(z_q - z) == z_q
        float d = zq - zv;
        acc += d * d;
        if (t == 0) {
            out[OUT_ONEHOT + (size_t)row * N_E + idx] = 1.0f;
            out[OUT_IDX + row] = (float)idx;
            atomicAdd(&counts[idx], 1u);
        }
    }

    red[t] = acc;
    __syncthreads();
    for (int s = 128; s > 0; s >>= 1) {
        if (t < s) red[t] += red[t + s];
        __syncthreads();
    }
    if (t == 0) atomicAdd(lossAcc, red[0]);
}

// ---------------------------------------------------------------------------
// Kernel 4: perplexity from counts + final loss scalar
// ---------------------------------------------------------------------------
__global__ __launch_bounds__(256) void vq_finalize(
    const unsigned int* __restrict__ counts, const float* __restrict__ lossAcc,
    float* __restrict__ out) {
    __shared__ float red[256];
    const int t = threadIdx.x;
    float s = 0.f;
    for (int c = t; c < N_E; c += 256) {
        float p = (float)counts[c] * (1.0f / (float)N_ROWS);
        s += p * logf(p + 1e-10f);
    }
    red[t] = s;
    __syncthreads();
    for (int st = 128; st > 0; st >>= 1) {
        if (t < st) red[t] += red[t + st];
        __syncthreads();
    }
    if (t == 0) {
        out[OUT_PERP] = expf(-red[0]);
        out[OUT_LOSS] = (1.0f + BETA) * (*lossAcc) / (float)(N_ROWS * E_DIM);
    }
}

// ---------------------------------------------------------------------------
extern "C" void kernel_launch(void* const* d_in, const int* in_sizes, int n_in,
                              void* d_out, int out_size, void* d_ws, size_t ws_size,
                              hipStream_t stream) {
    const float* z   = (const float*)d_in[0];
    const float* emb = (const float*)d_in[1];
    float* out = (float*)d_out;

    // ws layout: e2[2048] f32 | idx[32768] i32 | counts[2048] u32 | lossAcc f32
    char* ws = (char*)d_ws;
    float*        e2      = (float*)ws;
    int*          idxArr  = (int*)(ws + 8192);
    unsigned int* counts  = (unsigned int*)(ws + 8192 + 131072);
    float*        lossAcc = (float*)(ws + 8192 + 131072 + 8192);

    vq_init<<<8, 256, 0, stream>>>(emb, e2, counts, lossAcc);
    hipMemsetAsync(out + OUT_ONEHOT, 0, (size_t)N_ROWS * N_E * sizeof(float), stream);
    vq_argmin_wmma<<<N_ROWS / 64, 128, 0, stream>>>(z, emb, e2, idxArr);
    vq_outputs<<<N_ROWS / 32, 256, 0, stream>>>(z, emb, idxArr, out, counts, lossAcc);
    vq_finalize<<<1, 256, 0, stream>>>(counts, lossAcc, out);
}